// Node2Vec_24189255811345
// MI455X (gfx1250) — compile-verified
//
#include <hip/hip_runtime.h>
#include <hip/hip_bf16.h>

#define NUM_NODES 200000
#define NUM_EDGES 640000
#define RANK      16
#define EMBED_DIM 128

typedef __attribute__((ext_vector_type(2))) float v2f;
typedef __attribute__((ext_vector_type(8))) float v8f;

// ---------------------------------------------------------------------------
// Kernel 1: zero the workspace (accum [N,16] + counts [N]) — ws is poisoned.
// ---------------------------------------------------------------------------
__global__ __launch_bounds__(256) void zero_f32_kernel(float* __restrict__ p, int n) {
    int i = blockIdx.x * blockDim.x + threadIdx.x;
    if (i < n) p[i] = 0.0f;
}

// ---------------------------------------------------------------------------
// Kernel 2: rank-space scatter.  16 lanes cooperate on one edge: lane c adds
// U[src][c] into accum[dst][c] (64B coalesced gather per edge, 4B atomics).
// Lane 0 of each group increments the destination count.
// ---------------------------------------------------------------------------
__global__ __launch_bounds__(256) void scatter_rank_kernel(
    const float* __restrict__ U,
    const int*   __restrict__ edge_index,   // [2, E] row-major: src then dst
    float* __restrict__ accum,              // [N, RANK]
    float* __restrict__ counts)             // [N]
{
    long long tid = (long long)blockIdx.x * blockDim.x + threadIdx.x;
    int e = (int)(tid >> 4);
    int c = (int)(tid & 15);
    if (e >= NUM_EDGES) return;

    int s = edge_index[e];              // src row
    int d = edge_index[NUM_EDGES + e];  // dst row

    float val = U[(long long)s * RANK + c];
    atomicAdd(&accum[(long long)d * RANK + c], val);
    if (c == 0) atomicAdd(&counts[d], 1.0f);
}

// ---------------------------------------------------------------------------
// Kernel 3: out[m, :] = (accum[m, :] / max(count[m],1)) @ V   via
// V_WMMA_F32_16X16X4_F32 chained over K=16 (4 steps of K=4).
// Block = 256 threads = 8 waves; block handles 16 nodes; wave w handles
// output columns [16w, 16w+16).  V (16x128 f32 = 8KB) staged in LDS.
//
// A 16x4 f32 layout (ISA 7.12.2): lanes 0-15 -> M=lane, VGPRs = K{0,1};
// lanes 16-31 -> M=lane-16, VGPRs = K{2,3}.  B 4x16 mirrors (rows K split
// across lane halves).  C/D 16x16: VGPR i -> row i (lanes 0-15, N=lane) and
// row i+8 (lanes 16-31, N=lane-16).
// ---------------------------------------------------------------------------
__global__ __launch_bounds__(256) void lowrank_gemm_scale_kernel(
    const float* __restrict__ accum,    // [N, RANK]
    const float* __restrict__ counts,   // [N]
    const float* __restrict__ V,        // [RANK, EMBED_DIM] row-major
    float* __restrict__ out)            // [N, EMBED_DIM]
{
    __shared__ float sV[RANK * EMBED_DIM];   // 8 KB

    // Stage V into LDS: 2048 floats / 256 threads = 8 each.
    #pragma unroll
    for (int i = 0; i < 8; ++i)
        sV[threadIdx.x + i * 256] = V[threadIdx.x + i * 256];
    __syncthreads();

    const int lane   = threadIdx.x & 31;
    const int wave   = threadIdx.x >> 5;       // 0..7 -> column tile
    const int m0     = blockIdx.x * 16;        // node-tile base
    const int lhalf  = lane >> 4;              // 0: lanes 0-15, 1: lanes 16-31
    const int l15    = lane & 15;
    const int n0     = wave * 16;

    // Row this lane's A operand belongs to (row-uniform per lane -> fold the
    // 1/max(count,1) scaling into A before the matmul).
    const int m = m0 + l15;
    const float cnt = counts[m];
    const float inv = 1.0f / fmaxf(cnt, 1.0f);

    const float* arow = accum + (long long)m * RANK;
    const int kh = lhalf * 2;                  // k offset within each K=4 group

    v8f c = {};
    #pragma unroll
    for (int kb = 0; kb < RANK; kb += 4) {
        v2f a, b;
        a.x = arow[kb + kh + 0] * inv;
        a.y = arow[kb + kh + 1] * inv;
        const int n = n0 + l15;
        b.x = sV[(kb + kh + 0) * EMBED_DIM + n];
        b.y = sV[(kb + kh + 1) * EMBED_DIM + n];
        c = __builtin_amdgcn_wmma_f32_16x16x4_f32(
                /*neg_a=*/false, a, /*neg_b=*/false, b,
                /*c_mod=*/(short)0, c, /*reuse_a=*/false, /*reuse_b=*/false);
    }

    // Writeback: VGPR i -> rows m0+i (lane half 0) / m0+8+i (lane half 1),
    // column n0 + (lane&15).
    const int ncol  = n0 + l15;
    const int mbase = m0 + lhalf * 8;
    float* op = out + (long long)mbase * EMBED_DIM + ncol;
    #pragma unroll
    for (int i = 0; i < 8; ++i)
        op[(long long)i * EMBED_DIM] = c[i];
}

// ---------------------------------------------------------------------------
extern "C" void kernel_launch(void* const* d_in, const int* in_sizes, int n_in,
                              void* d_out, int out_size, void* d_ws, size_t ws_size,
                              hipStream_t stream) {
    const float* U          = (const float*)d_in[0];   // [N, 16]
    const float* V          = (const float*)d_in[1];   // [16, 128]
    const int*   edge_index = (const int*)d_in[2];     // [2, E]
    float* out = (float*)d_out;                        // [N, 128]

    float* accum  = (float*)d_ws;                      // [N, 16]
    float* counts = accum + (size_t)NUM_NODES * RANK;  // [N]

    // 1) zero accum + counts
    {
        int n = NUM_NODES * RANK + NUM_NODES;
        int blocks = (n + 255) / 256;
        zero_f32_kernel<<<blocks, 256, 0, stream>>>(accum, n);
    }
    // 2) scatter U[src] into accum[dst] in rank space
    {
        long long threads = (long long)NUM_EDGES * 16;
        int blocks = (int)((threads + 255) / 256);
        scatter_rank_kernel<<<blocks, 256, 0, stream>>>(U, edge_index, accum, counts);
    }
    // 3) WMMA low-rank GEMM with folded mean-scaling
    {
        int blocks = NUM_NODES / 16;   // 12500, N divisible by 16
        lowrank_gemm_scale_kernel<<<blocks, 256, 0, stream>>>(accum, counts, V, out);
    }
}